// WangGovindCarter99_71150428226212
// MI455X (gfx1250) — compile-verified
//
#include <hip/hip_runtime.h>
#include <math.h>

// ---------------------------------------------------------------------------
// WGC99 KEDF on 192^3 grid for MI455X (gfx1250).
// 3D FFTs implemented as separable 192-point DFT matmuls on the f32 WMMA
// pipe (V_WMMA_F32_16X16X4_F32), with axis rotation between passes.
// ---------------------------------------------------------------------------

#define NDIM   192
#define NPTS   (192*192*192)
#define NLINES (192*192)
#define NETA   2000
#define ETAMAX 60.0f
#define CTF    2.871234000188191f   // 0.3*(3*pi^2)^(2/3)
#define DPI    3.14159265358979323846

typedef float v2f __attribute__((ext_vector_type(2)));
typedef float v8f __attribute__((ext_vector_type(8)));

__device__ __forceinline__ v8f wmma_f32(v2f a, v2f b, v8f c) {
    // D(16x16) = A(16x4) * B(4x16) + C ; fp32 matrix core
    return __builtin_amdgcn_wmma_f32_16x16x4_f32(
        /*neg_a=*/false, a, /*neg_b=*/false, b,
        /*c_mod=*/(short)0, c, /*reuse_a=*/false, /*reuse_b=*/false);
}

// ---------------------------------------------------------------------------
// Mean reduction (deterministic, two-stage, no float atomics)
// ---------------------------------------------------------------------------
__global__ __launch_bounds__(256)
void reduce_partial_kernel(const float* __restrict__ den, float* __restrict__ part) {
    __shared__ float sm[256];
    float s = 0.f;
    for (size_t i = (size_t)blockIdx.x * 256 + threadIdx.x; i < (size_t)NPTS;
         i += (size_t)gridDim.x * 256)
        s += den[i];
    sm[threadIdx.x] = s;
    __syncthreads();
    for (int o = 128; o > 0; o >>= 1) {
        if ((int)threadIdx.x < o) sm[threadIdx.x] += sm[threadIdx.x + o];
        __syncthreads();
    }
    if (threadIdx.x == 0) part[blockIdx.x] = sm[0];
}

__global__ __launch_bounds__(256)
void finalize_scalars_kernel(const float* __restrict__ part, int nPart,
                             const float* __restrict__ kappa,
                             float* __restrict__ scal) {
    __shared__ float sm[256];
    float s = 0.f;
    for (int i = threadIdx.x; i < nPart; i += 256) s += part[i];
    sm[threadIdx.x] = s;
    __syncthreads();
    for (int o = 128; o > 0; o >>= 1) {
        if ((int)threadIdx.x < o) sm[threadIdx.x] += sm[threadIdx.x + o];
        __syncthreads();
    }
    if (threadIdx.x == 0) {
        float mean = sm[0] / (float)NPTS;
        float rho0 = kappa[0] * mean;
        float kF   = cbrtf(3.0f * (float)(DPI * DPI) * rho0);
        scal[0] = mean;
        scal[1] = rho0;
        scal[2] = kF;
        scal[3] = 0.5f / kF;    // 1/(2 kF)
        scal[4] = 1.0f / rho0;
    }
}

// ---------------------------------------------------------------------------
// Lindhard kernel table + jnp.gradient-style derivative tables
// ---------------------------------------------------------------------------
__device__ __forceinline__ float lindhard_w0(float eta, float a, float b) {
    float e = eta < 1e-8f ? 1e-8f : eta;
    bool near1 = fabsf(e - 1.0f) < 1e-8f;
    float denom = near1 ? 1.0f : (1.0f - e);
    float ratio = fabsf((1.0f + e) / denom);
    float G = 0.5f + (near1 ? 0.0f : (1.0f - e * e) / (4.0f * e) * logf(ratio));
    float w = 1.0f / G - 3.0f * e * e - 1.0f;
    if (eta < 1e-8f) w = 0.0f;
    return (5.0f / (9.0f * a * b)) * w;
}

__global__ void build_w0_kernel(float* __restrict__ w0t,
                                const float* __restrict__ alpha,
                                const float* __restrict__ beta, float h) {
    int i = blockIdx.x * blockDim.x + threadIdx.x;
    if (i >= NETA) return;
    w0t[i] = lindhard_w0((float)i * h, alpha[0], beta[0]);
}

__global__ void grad_kernel(const float* __restrict__ src, float* __restrict__ dst,
                            int n, float inv_h) {
    int i = blockIdx.x * blockDim.x + threadIdx.x;
    if (i >= n) return;
    float g = (i == 0)     ? (src[1] - src[0])
            : (i == n - 1) ? (src[n - 1] - src[n - 2])
                           : 0.5f * (src[i + 1] - src[i - 1]);
    dst[i] = g * inv_h;
}

// ---------------------------------------------------------------------------
// 192x192 forward / inverse DFT matrices (fp64 twiddle gen, exact integer phase)
// ---------------------------------------------------------------------------
__global__ void build_dft_kernel(float* __restrict__ Wfre, float* __restrict__ Wfim,
                                 float* __restrict__ Wire, float* __restrict__ Wiim) {
    int idx = blockIdx.x * blockDim.x + threadIdx.x;
    if (idx >= NDIM * NDIM) return;
    int k = idx / NDIM, n = idx % NDIM;
    int m = (k * n) % NDIM;
    double ang = -2.0 * DPI * (double)m / (double)NDIM;
    double c = cos(ang), s = sin(ang);
    Wfre[idx] = (float)c;                    // e^{-i 2pi kn/N}
    Wfim[idx] = (float)s;
    Wire[idx] = (float)(c / (double)NDIM);   // e^{+i 2pi kn/N} / N  (1/N per axis)
    Wiim[idx] = (float)(-s / (double)NDIM);
}

// ---------------------------------------------------------------------------
// Pointwise prep: 5 real fields + n^alpha plane
// ---------------------------------------------------------------------------
__global__ __launch_bounds__(256)
void prep_kernel(const float* __restrict__ den, const float* __restrict__ aP,
                 const float* __restrict__ bP, const float* __restrict__ scal,
                 float* __restrict__ f0, float* __restrict__ f1,
                 float* __restrict__ f2, float* __restrict__ f3,
                 float* __restrict__ f4, float* __restrict__ naP) {
    size_t i = (size_t)blockIdx.x * 256 + threadIdx.x;
    if (i >= (size_t)NPTS) return;
    float a = aP[0], b = bP[0], rho0 = scal[1];
    float d  = den[i];
    float sq = d > 0.f ? sqrtf(d) : 0.f;
    float nb = powf(d, b);
    float na = powf(d, a);
    float th = d - rho0;
    f0[i] = d;
    f1[i] = sq;
    f2[i] = nb;
    f3[i] = th * nb;
    f4[i] = th * th * nb;
    naP[i] = na;
}

// ---------------------------------------------------------------------------
// DFT pass along innermost axis with transpose-on-store (axis rotation).
// One wave per block: 16 lines x full 192-point DFT via f32 WMMA.
//   in : [field][line(0..NLINES-1)][n(0..191)]
//   out: [field][n_out(0..191)][line]
// ---------------------------------------------------------------------------
template <bool REAL_IN, bool REAL_OUT>
__global__ __launch_bounds__(32)
void dft_pass_kernel(const float* __restrict__ inRe, const float* __restrict__ inIm,
                     float* __restrict__ outRe, float* __restrict__ outIm,
                     const float* __restrict__ Wre, const float* __restrict__ Wim,
                     size_t planeStride) {
    __shared__ float sRe[NDIM * 16];
    __shared__ float sIm[NDIM * 16];

    const int lane = threadIdx.x;
    const int lm   = lane & 15;   // M index (A rows / C cols)
    const int half = lane >> 4;   // K-half select
    const size_t fbase    = (size_t)blockIdx.y * planeStride;
    const size_t lineBase = (size_t)blockIdx.x * 16;

    // Stage 16 lines x 192 values into LDS, transposed: s[n*16 + line]
    #pragma unroll 4
    for (int t = 0; t < 24; ++t) {
        int idx  = t * 32 + lane;     // 768 float4 chunks total
        int line = idx / 48;
        int q    = idx % 48;
        const size_t gi = fbase + (lineBase + (size_t)line) * NDIM + (size_t)q * 4;
        float4 vr = *(const float4*)&inRe[gi];
        sRe[(q * 4 + 0) * 16 + line] = vr.x;
        sRe[(q * 4 + 1) * 16 + line] = vr.y;
        sRe[(q * 4 + 2) * 16 + line] = vr.z;
        sRe[(q * 4 + 3) * 16 + line] = vr.w;
        if constexpr (!REAL_IN) {
            float4 vi = *(const float4*)&inIm[gi];
            sIm[(q * 4 + 0) * 16 + line] = vi.x;
            sIm[(q * 4 + 1) * 16 + line] = vi.y;
            sIm[(q * 4 + 2) * 16 + line] = vi.z;
            sIm[(q * 4 + 3) * 16 + line] = vi.w;
        }
    }
    __syncthreads();

    // 12 output tiles of 16 frequencies
    for (int t = 0; t < NDIM / 16; ++t) {
        v8f RR = {0,0,0,0,0,0,0,0};
        v8f II = {0,0,0,0,0,0,0,0};
        v8f RI = {0,0,0,0,0,0,0,0};
        v8f IR = {0,0,0,0,0,0,0,0};

        const int Mrow = t * 16 + lm;                 // output frequency (A row)
        const float* wr = &Wre[(size_t)Mrow * NDIM];
        const float* wi = &Wim[(size_t)Mrow * NDIM];

        for (int kk = 0; kk < NDIM; kk += 4) {
            // A operand (16x4): lane<16 holds K=kk,kk+1 ; lane>=16 holds K=kk+2,kk+3
            int ac = kk + half * 2;
            v2f ar = *(const v2f*)&wr[ac];
            v2f ai = *(const v2f*)&wi[ac];
            // B operand (4x16): same K split, N = lm, from LDS
            int b0 = (kk + half * 2) * 16 + lm;
            v2f br; br.x = sRe[b0]; br.y = sRe[b0 + 16];
            RR = wmma_f32(ar, br, RR);
            IR = wmma_f32(ai, br, IR);
            if constexpr (!REAL_IN) {
                v2f bi; bi.x = sIm[b0]; bi.y = sIm[b0 + 16];
                RI = wmma_f32(ar, bi, RI);
                II = wmma_f32(ai, bi, II);
            }
        }

        v8f Dr, Di;
        if constexpr (REAL_IN) { Dr = RR;      Di = IR; }
        else                   { Dr = RR - II; Di = RI + IR; }

        // C/D layout: VGPR r, lanes0-15 -> M=r, lanes16-31 -> M=r+8 ; N=lm
        #pragma unroll
        for (int r = 0; r < 8; ++r) {
            size_t o = fbase + (size_t)(t * 16 + r + half * 8) * NLINES + lineBase + lm;
            outRe[o] = Dr[r];
            if constexpr (!REAL_OUT) outIm[o] = Di[r];
        }
    }
}

// ---------------------------------------------------------------------------
// Spectral multipliers: -k^2 on den/sqrt(den); Lindhard combos on nb fields.
// Groups the six convolutions into three inverse transforms:
//   S_A = K0*F(nb) + (K1/r0)*F(tnb) + 0.5*(K2s/r0^2)*F(ttnb)   (coeff 1)
//   S_B = (K1/r0)*F(nb) + (K2m/r0^2)*F(tnb)                    (coeff theta)
//   S_C = (K2s/r0^2)*F(nb)                                     (coeff theta^2/2)
// ---------------------------------------------------------------------------
__global__ __launch_bounds__(256)
void spectral_kernel(const float* __restrict__ kxyz, const float* __restrict__ scal,
                     const float* __restrict__ w0t, const float* __restrict__ dwt,
                     const float* __restrict__ d2wt,
                     float* __restrict__ Yre, float* __restrict__ Yim, size_t plane) {
    size_t idx = (size_t)blockIdx.x * 256 + threadIdx.x;
    if (idx >= (size_t)NPTS) return;
    int kkk = (int)(idx % NDIM);
    int j   = (int)((idx / NDIM) % NDIM);
    int i   = (int)(idx / (NDIM * NDIM));
    int kz  = (kkk <= NDIM / 2) ? kkk : (NDIM - kkk);   // rfft mirror (k^2 even)
    size_t kid = (((size_t)i * NDIM + j) * 97 + kz) * 3;
    float kx = kxyz[kid], ky = kxyz[kid + 1], kzv = kxyz[kid + 2];
    float k2 = kx * kx + ky * ky + kzv * kzv;

    float eta = sqrtf(k2) * scal[3];
    const float h = ETAMAX / (float)(NETA - 1);
    float p = eta / h;
    if (p < 0.f) p = 0.f;
    int i0 = (int)p;
    if (i0 > NETA - 2) i0 = NETA - 2;
    float fr = p - (float)i0;
    if (fr > 1.f) fr = 1.f;
    float K0  = w0t[i0]  + (w0t[i0 + 1]  - w0t[i0])  * fr;
    float Kd  = dwt[i0]  + (dwt[i0 + 1]  - dwt[i0])  * fr;
    float Kdd = d2wt[i0] + (d2wt[i0 + 1] - d2wt[i0]) * fr;

    float K1  = -(eta * Kd) / 6.0f;
    float K2m = (eta * eta * Kdd + eta * Kd) / 36.0f;
    float K2s = (eta * eta * Kdd + 7.0f * eta * Kd) / 36.0f;

    float invr = scal[4];
    float c1  = K1 * invr;
    float c2s = K2s * invr * invr;
    float c2m = K2m * invr * invr;

    float dR  = Yre[idx],              dI  = Yim[idx];
    float sR  = Yre[plane + idx],      sI  = Yim[plane + idx];
    float nbR = Yre[2 * plane + idx],  nbI = Yim[2 * plane + idx];
    float tR  = Yre[3 * plane + idx],  tI  = Yim[3 * plane + idx];
    float ttR = Yre[4 * plane + idx],  ttI = Yim[4 * plane + idx];

    float nk2 = -k2;
    Yre[idx] = nk2 * dR;              Yim[idx] = nk2 * dI;
    Yre[plane + idx] = nk2 * sR;      Yim[plane + idx] = nk2 * sI;
    Yre[2 * plane + idx] = K0 * nbR + c1 * tR + 0.5f * c2s * ttR;
    Yim[2 * plane + idx] = K0 * nbI + c1 * tI + 0.5f * c2s * ttI;
    Yre[3 * plane + idx] = c1 * nbR + c2m * tR;
    Yim[3 * plane + idx] = c1 * nbI + c2m * tI;
    Yre[4 * plane + idx] = c2s * nbR;
    Yim[4 * plane + idx] = c2s * nbI;
}

// ---------------------------------------------------------------------------
// Final assembly: tau = TF + vW + nonlocal
// ---------------------------------------------------------------------------
__global__ __launch_bounds__(256)
void final_kernel(const float* __restrict__ den, const float* __restrict__ naP,
                  const float* __restrict__ scal, const float* __restrict__ Xre,
                  size_t plane, float* __restrict__ out) {
    size_t i = (size_t)blockIdx.x * 256 + threadIdx.x;
    if (i >= (size_t)NPTS) return;
    float rho0 = scal[1];
    float d  = den[i];
    float sq = d > 0.f ? sqrtf(d) : 0.f;
    float th = d - rho0;
    float lapd = Xre[i];
    float laps = Xre[plane + i];
    float A = Xre[2 * plane + i];
    float B = Xre[3 * plane + i];
    float C = Xre[4 * plane + i];
    float tau_tf = CTF * powf(d, 5.0f / 3.0f);
    float tau_vw = 0.25f * lapd - 0.5f * sq * laps;
    float tau_nl = CTF * naP[i] * (A + th * B + 0.5f * th * th * C);
    out[i] = tau_tf + tau_vw + tau_nl;
}

// ---------------------------------------------------------------------------
// Host launcher
// ---------------------------------------------------------------------------
extern "C" void kernel_launch(void* const* d_in, const int* in_sizes, int n_in,
                              void* d_out, int out_size, void* d_ws, size_t ws_size,
                              hipStream_t stream) {
    (void)in_sizes; (void)n_in; (void)out_size; (void)ws_size;
    const float* den   = (const float*)d_in[0];
    const float* kxyz  = (const float*)d_in[1];
    const float* alpha = (const float*)d_in[2];
    const float* beta  = (const float*)d_in[3];
    const float* kappa = (const float*)d_in[5];   // gamma (d_in[4]) unused by reference
    float* out = (float*)d_out;

    const size_t P = (size_t)NPTS;
    float* ws   = (float*)d_ws;
    float* part = ws;                          // 1024
    float* scal = ws + 1024;                   // 16
    float* w0t  = ws + 1040;                   // NETA
    float* dwt  = w0t + NETA;
    float* d2wt = dwt + NETA;
    float* Wfre = ws + 8192;                   // 4 * 192*192 DFT matrices
    float* Wfim = Wfre + NDIM * NDIM;
    float* Wire = Wfim + NDIM * NDIM;
    float* Wiim = Wire + NDIM * NDIM;
    float* naP  = ws + 8192 + 4 * (size_t)(NDIM * NDIM);
    float* R    = naP + P;                     // 5 real prep planes
    float* Xre  = R + 5 * P;                   // ping complex (5 planes re/im)
    float* Xim  = Xre + 5 * P;
    float* Yre  = Xim + 5 * P;                 // pong complex
    float* Yim  = Yre + 5 * P;

    const float h = ETAMAX / (float)(NETA - 1);
    const int ptBlocks = (int)((P + 255) / 256);

    // Scalars + tables + DFT matrices
    reduce_partial_kernel<<<864, 256, 0, stream>>>(den, part);
    finalize_scalars_kernel<<<1, 256, 0, stream>>>(part, 864, kappa, scal);
    build_w0_kernel<<<(NETA + 255) / 256, 256, 0, stream>>>(w0t, alpha, beta, h);
    grad_kernel<<<(NETA + 255) / 256, 256, 0, stream>>>(w0t, dwt, NETA, 1.0f / h);
    grad_kernel<<<(NETA + 255) / 256, 256, 0, stream>>>(dwt, d2wt, NETA, 1.0f / h);
    build_dft_kernel<<<(NDIM * NDIM + 255) / 256, 256, 0, stream>>>(Wfre, Wfim, Wire, Wiim);

    // Real fields
    prep_kernel<<<ptBlocks, 256, 0, stream>>>(den, alpha, beta, scal,
                                              R, R + P, R + 2 * P, R + 3 * P, R + 4 * P, naP);

    // Forward 3D DFT (3 rotating passes), 5 fields batched in grid.y
    dim3 dgrid(NLINES / 16, 5);
    dft_pass_kernel<true,  false><<<dgrid, 32, 0, stream>>>(R,   nullptr, Yre, Yim, Wfre, Wfim, P);
    dft_pass_kernel<false, false><<<dgrid, 32, 0, stream>>>(Yre, Yim,     Xre, Xim, Wfre, Wfim, P);
    dft_pass_kernel<false, false><<<dgrid, 32, 0, stream>>>(Xre, Xim,     Yre, Yim, Wfre, Wfim, P);

    // Spectral multipliers (in-place on Y)
    spectral_kernel<<<ptBlocks, 256, 0, stream>>>(kxyz, scal, w0t, dwt, d2wt, Yre, Yim, P);

    // Inverse 3D DFT (last pass stores real part only into Xre planes)
    dft_pass_kernel<false, false><<<dgrid, 32, 0, stream>>>(Yre, Yim, Xre, Xim,     Wire, Wiim, P);
    dft_pass_kernel<false, false><<<dgrid, 32, 0, stream>>>(Xre, Xim, Yre, Yim,     Wire, Wiim, P);
    dft_pass_kernel<false, true ><<<dgrid, 32, 0, stream>>>(Yre, Yim, Xre, nullptr, Wire, Wiim, P);

    // Assemble tau
    final_kernel<<<ptBlocks, 256, 0, stream>>>(den, naP, scal, Xre, P, out);
}